// AttributeDecoder_11570641896117
// MI455X (gfx1250) — compile-verified
//
#include <hip/hip_runtime.h>

// GCN 2-layer forward for MI455X (gfx1250, wave32).
// Dense GEMMs: V_WMMA_F32_16X16X4_F32, B-tile register-resident, compile-time
// strides so the inner loop is global_load(imm offset) + wmma only.
// Sparse aggregation: per-edge norm precomputed once (shared by both layers,
// like the reference), then float4 gather + f32 atomic scatter-add.

#define NN 50000      // nodes (== 3125 * 16, exact WMMA tiling; 3125 = 625*5)
#define NE 800000     // edges
#define NH 64         // hidden
#define NF 128        // out features
#define MTILES 5      // row tiles per wave in the GEMM (divides 3125)

typedef float v2f __attribute__((ext_vector_type(2)));
typedef float v8f __attribute__((ext_vector_type(8)));

// ---------------- utility kernels ----------------

__global__ void k_fill(float* __restrict__ p, float v, int n) {
    int i = blockIdx.x * blockDim.x + threadIdx.x;
    if (i < n) p[i] = v;
}

__global__ void k_fill4(float4* __restrict__ p, float v, int n4) {
    int i = blockIdx.x * blockDim.x + threadIdx.x;
    if (i < n4) p[i] = make_float4(v, v, v, v);
}

__global__ void k_deg_scatter(const int* __restrict__ dst, float* __restrict__ deg, int ne) {
    int e = blockIdx.x * blockDim.x + threadIdx.x;
    if (e < ne) atomicAdd(&deg[dst[e]], 1.0f);
}

__global__ void k_rsqrt(float* __restrict__ d, int n) {
    int i = blockIdx.x * blockDim.x + threadIdx.x;
    if (i < n) d[i] = rsqrtf(d[i]);   // deg >= 1 always (self loop present)
}

// Per-edge symmetric norm, computed once and reused by both layers:
// w[e] = dinv[src[e]] * dinv[dst[e]]
__global__ void k_edge_norm(const int* __restrict__ src, const int* __restrict__ dst,
                            const float* __restrict__ dinv, float* __restrict__ w, int ne) {
    int e = blockIdx.x * blockDim.x + threadIdx.x;
    if (e < ne) w[e] = dinv[src[e]] * dinv[dst[e]];
}

// ---------------- fp32 WMMA GEMM ----------------
// C[M x NC] = A[M x 64] * B[64 x NC], row-major. NC is compile-time so all B
// strides fold into immediate offsets. One wave owns one 16-column tile and
// keeps the whole 64x16 B tile in 32 VGPRs across MTILES row tiles.
// Block = (NC/16) waves; blockIdx.x = group of MTILES row tiles.
// No divergence anywhere -> EXEC all-1s as WMMA requires.
template <int NC>
__global__ void k_gemm_f32_wmma(const float* __restrict__ A,
                                const float* __restrict__ B,
                                float* __restrict__ C) {
    const int lane    = threadIdx.x & 31;
    const int wave    = threadIdx.x >> 5;
    const int half    = lane >> 4;    // 0: lanes 0-15, 1: lanes 16-31
    const int l16     = lane & 15;
    const int colBase = wave << 4;

    // --- load B tile into registers (16 k-steps of v2f) ---
    // B 4x16 f32 operand layout: VGPR0 = row K=k0+2*half, VGPR1 = row +1.
    const float* bp = B + colBase + l16;
    v2f breg[16];
#pragma unroll
    for (int kk = 0; kk < 16; ++kk) {
        const int ka = kk * 4 + 2 * half;
        breg[kk].x = bp[ka * NC];           // compile-time immediate offsets
        breg[kk].y = bp[(ka + 1) * NC];
    }

    const int rowBase0 = blockIdx.x * (MTILES * 16);

#pragma unroll
    for (int mt = 0; mt < MTILES; ++mt) {
        const int rowBase = rowBase0 + mt * 16;
        // A 16x4 f32 operand layout: lane half selects K pair; contiguous 8B.
        const float* arow = A + (size_t)(rowBase + l16) * 64 + 2 * half;

        v8f acc = {0.f, 0.f, 0.f, 0.f, 0.f, 0.f, 0.f, 0.f};
#pragma unroll
        for (int kk = 0; kk < 16; ++kk) {
            v2f a = *(const v2f*)(arow + kk * 4);   // global_load_b64, imm offset
            acc = __builtin_amdgcn_wmma_f32_16x16x4_f32(
                false, a, false, breg[kk], (short)0, acc, false, false);
        }

        // C/D 16x16 f32 layout: VGPR r -> row r (lanes 0-15) / r+8 (lanes 16-31)
        float* crow = C + (size_t)(rowBase + (half << 3)) * NC + colBase + l16;
#pragma unroll
        for (int r = 0; r < 8; ++r)
            crow[r * NC] = acc[r];
    }
}

// ---------------- edge scatter-add (float4) ----------------
// One thread per (edge, 4 features): sequential w[e] read (precomputed norm),
// float4 gather of h[src], 4 f32 atomic adds into destination row.
template <int LOGF>
__global__ void k_edge_scatter4(const int* __restrict__ src, const int* __restrict__ dst,
                                const float* __restrict__ wnorm, const float* __restrict__ h,
                                float* __restrict__ out) {
    constexpr int LOGQ = LOGF - 2;                 // quads per row = 1<<LOGQ
    const int idx = blockIdx.x * blockDim.x + threadIdx.x;
    if (idx >= (NE << LOGQ)) return;
    const int e  = idx >> LOGQ;
    const int q  = idx & ((1 << LOGQ) - 1);
    const int s  = src[e];
    const int d  = dst[e];
    const float w = wnorm[e];
    const float4 v = *(const float4*)(h + (s << LOGF) + (q << 2));
    float* o = out + (d << LOGF) + (q << 2);
    atomicAdd(o + 0, v.x * w);
    atomicAdd(o + 1, v.y * w);
    atomicAdd(o + 2, v.z * w);
    atomicAdd(o + 3, v.w * w);
}

// ---------------- fused self-loop + bias + ReLU (float4) ----------------
// acc[n,f] = relu(acc[n,f] + dinv[n]^2 * hself[n,f] + bias[f])
template <int LOGF>
__global__ void k_finish4(float* __restrict__ acc, const float* __restrict__ hself,
                          const float* __restrict__ dinv, const float* __restrict__ bias) {
    constexpr int LOGQ = LOGF - 2;
    const int i = blockIdx.x * blockDim.x + threadIdx.x;
    if (i >= (NN << LOGQ)) return;
    const int n = i >> LOGQ;
    const int q = i & ((1 << LOGQ) - 1);
    const float di = dinv[n];
    const float w  = di * di;
    const float4 a = ((const float4*)acc)[i];
    const float4 hs = ((const float4*)hself)[i];
    const float4 b  = ((const float4*)bias)[q];
    float4 r;
    r.x = a.x + w * hs.x + b.x;  r.x = r.x > 0.f ? r.x : 0.f;
    r.y = a.y + w * hs.y + b.y;  r.y = r.y > 0.f ? r.y : 0.f;
    r.z = a.z + w * hs.z + b.z;  r.z = r.z > 0.f ? r.z : 0.f;
    r.w = a.w + w * hs.w + b.w;  r.w = r.w > 0.f ? r.w : 0.f;
    ((float4*)acc)[i] = r;
}

// ---------------- host launch ----------------

extern "C" void kernel_launch(void* const* d_in, const int* in_sizes, int n_in,
                              void* d_out, int out_size, void* d_ws, size_t ws_size,
                              hipStream_t stream) {
    (void)in_sizes; (void)n_in; (void)out_size; (void)ws_size;

    const float* x   = (const float*)d_in[0];
    const int*   ei  = (const int*)d_in[1];    // [2, NE] flat
    const int*   src = ei;
    const int*   dst = ei + NE;
    const float* W1  = (const float*)d_in[2];
    const float* b1  = (const float*)d_in[3];
    const float* W2  = (const float*)d_in[4];
    const float* b2  = (const float*)d_in[5];
    float* out = (float*)d_out;

    float* ws    = (float*)d_ws;
    float* dinv  = ws;                             // NN floats (deg -> rsqrt in place)
    float* wnorm = dinv + NN;                      // NE floats (per-edge norm)
    float* h1    = wnorm + NE;                     // NN*64  (16B-aligned offsets)
    float* acc1  = h1 + (size_t)NN * NH;           // NN*64  (layer-1 output after finish)
    float* g2    = acc1 + (size_t)NN * NH;         // NN*128

    const int T = 256;

    // ---- symmetric normalization: deg (with self loops) -> dinv -> per-edge w
    k_fill<<<(NN + T - 1) / T, T, 0, stream>>>(dinv, 1.0f, NN);
    k_deg_scatter<<<(NE + T - 1) / T, T, 0, stream>>>(dst, dinv, NE);
    k_rsqrt<<<(NN + T - 1) / T, T, 0, stream>>>(dinv, NN);
    k_edge_norm<<<(NE + T - 1) / T, T, 0, stream>>>(src, dst, dinv, wnorm, NE);

    // ---- layer 1: h1 = x @ W1 (WMMA), aggregate, relu(+b1) in place into acc1
    k_gemm_f32_wmma<NH><<<NN / 16 / MTILES, 4 * 32, 0, stream>>>(x, W1, h1);
    k_fill4<<<(NN * NH / 4 + T - 1) / T, T, 0, stream>>>((float4*)acc1, 0.0f, NN * NH / 4);
    k_edge_scatter4<6><<<((NE * (NH / 4)) + T - 1) / T, T, 0, stream>>>(src, dst, wnorm, h1, acc1);
    k_finish4<6><<<((NN * (NH / 4)) + T - 1) / T, T, 0, stream>>>(acc1, h1, dinv, b1);

    // ---- layer 2: g2 = acc1 @ W2 (WMMA), aggregate into d_out, relu(+b2)
    k_gemm_f32_wmma<NF><<<NN / 16 / MTILES, 8 * 32, 0, stream>>>(acc1, W2, g2);
    k_fill4<<<(NN * NF / 4 + T - 1) / T, T, 0, stream>>>((float4*)out, 0.0f, NN * NF / 4);
    k_edge_scatter4<7><<<((NE * (NF / 4)) + T - 1) / T, T, 0, stream>>>(src, dst, wnorm, g2, out);
    k_finish4<7><<<((NN * (NF / 4)) + T - 1) / T, T, 0, stream>>>(out, g2, dinv, b2);
}